// V3_GCN_SELF_ATTN_20023137534035
// MI455X (gfx1250) — compile-verified
//
#include <hip/hip_runtime.h>
#include <hip/hip_bf16.h>

typedef __attribute__((ext_vector_type(2))) float v2f;
typedef __attribute__((ext_vector_type(8))) float v8f;

#define NTOT 8192
#define NGRAPH 32
#define NODES 256
#define DKD 512
#define HCH 32
#define EDGES 65536
#define VBODY 2048
#define VFACE 512

__device__ __forceinline__ float wave_reduce_sum(float v) {
  #pragma unroll
  for (int off = 16; off >= 1; off >>= 1)
    v += __shfl_xor(v, off, 32);
  return v;
}

__device__ __forceinline__ v8f wmma_f32(v2f a, v2f b, v8f c) {
  return __builtin_amdgcn_wmma_f32_16x16x4_f32(
      /*neg_a=*/false, a, /*neg_b=*/false, b,
      /*c_mod=*/(short)0, c, /*reuse_a=*/false, /*reuse_b=*/false);
}

// ---------------------------------------------------------------------------
// Kernel 1: per-graph fused attention (rank-2 collapsed) + LayerNorm (analytic
// stats) + PReLU + Linear(512->32) via fp32 WMMA.  One block per graph.
// ---------------------------------------------------------------------------
__global__ __launch_bounds__(256)
void attn_mlp_kernel(const float* __restrict__ x,
                     const float* __restrict__ Wq, const float* __restrict__ Wk,
                     const float* __restrict__ Wv,
                     const float* __restrict__ ln_g, const float* __restrict__ ln_b,
                     const float* __restrict__ prelu_a,
                     const float* __restrict__ mlp_W, const float* __restrict__ mlp_b,
                     float* __restrict__ h_out) {
  __shared__ float sWv0[DKD], sWv1[DKD], sg[DKD], sb[DKD];
  __shared__ float sx0[NODES], sx1[NODES];
  __shared__ float sy0[NODES], sy1[NODES];
  __shared__ float smu[NODES], srstd[NODES];
  __shared__ float sred[NODES];
  __shared__ float sres[9];

  const int tid = threadIdx.x;
  const int nb = blockIdx.x * NODES;

  for (int d = tid; d < DKD; d += NODES) {
    sWv0[d] = Wv[d];
    sWv1[d] = Wv[DKD + d];
    sg[d]   = ln_g[d];
    sb[d]   = ln_b[d];
  }
  sx0[tid] = x[(nb + tid) * 2 + 0];
  sx1[tid] = x[(nb + tid) * 2 + 1];

  // Partials: M = Wq*Wk^T (2x2), Wv row means, Wv second moments.
  float p[9];
  #pragma unroll
  for (int q = 0; q < 9; q++) p[q] = 0.f;
  for (int d = tid; d < DKD; d += NODES) {
    float q0 = Wq[d], q1 = Wq[DKD + d];
    float k0 = Wk[d], k1 = Wk[DKD + d];
    float v0 = sWv0[d], v1 = sWv1[d];
    p[0] += q0 * k0; p[1] += q0 * k1; p[2] += q1 * k0; p[3] += q1 * k1;
    p[4] += v0;      p[5] += v1;
    p[6] += v0 * v0; p[7] += v0 * v1; p[8] += v1 * v1;
  }
  for (int q = 0; q < 9; q++) {
    __syncthreads();
    sred[tid] = p[q];
    __syncthreads();
    for (int s = NODES / 2; s > 0; s >>= 1) {
      if (tid < s) sred[tid] += sred[tid + s];
      __syncthreads();
    }
    if (tid == 0) sres[q] = sred[0];
  }
  __syncthreads();
  const float M00 = sres[0], M01 = sres[1], M10 = sres[2], M11 = sres[3];
  const float inv512 = 1.f / 512.f;
  const float mv0 = sres[4] * inv512, mv1 = sres[5] * inv512;
  const float s00 = sres[6] * inv512, s01 = sres[7] * inv512, s11 = sres[8] * inv512;

  // Softmax over rank-2 scores; y = softmax(S) @ x  ([256,2]).
  {
    const float scale = 0.04419417382415922f;  // 1/sqrt(512)
    float xi0 = sx0[tid], xi1 = sx1[tid];
    float ai = (xi0 * M00 + xi1 * M10) * scale;
    float bi = (xi0 * M01 + xi1 * M11) * scale;
    float mx = -3.4e38f;
    for (int j = 0; j < NODES; j++)
      mx = fmaxf(mx, ai * sx0[j] + bi * sx1[j]);
    float Z = 0.f, y0 = 0.f, y1 = 0.f;
    for (int j = 0; j < NODES; j++) {
      float e = __expf(ai * sx0[j] + bi * sx1[j] - mx);
      Z += e; y0 += e * sx0[j]; y1 += e * sx1[j];
    }
    float inv = 1.f / Z;
    y0 *= inv; y1 *= inv;
    sy0[tid] = y0; sy1[tid] = y1;
    // Analytic LayerNorm stats of out = y0*Wv0 + y1*Wv1.
    float mu  = y0 * mv0 + y1 * mv1;
    float ex2 = y0 * y0 * s00 + 2.f * y0 * y1 * s01 + y1 * y1 * s11;
    smu[tid]   = mu;
    srstd[tid] = rsqrtf(ex2 - mu * mu + 1e-5f);
  }
  __syncthreads();

  // WMMA f32 16x16x4: [256,512] @ [512,32], A computed on the fly.
  const float alpha = prelu_a[0];
  const int wave = tid >> 5, lane = tid & 31;
  const int ksel = (lane < 16) ? 0 : 2;
  const int mrow = lane & 15;
  const int n0 = lane & 15, n1 = n0 + 16;
  const float bias0 = mlp_b[n0], bias1 = mlp_b[n1];

  for (int mt = wave * 2; mt < wave * 2 + 2; mt++) {
    const int m = mt * 16 + mrow;
    const float y0m = sy0[m], y1m = sy1[m];
    const float mum = smu[m], rsm = srstd[m];
    v8f c0 = {}; v8f c1 = {};
    for (int k = 0; k < DKD; k += 4) {
      int d0 = k + ksel, d1 = d0 + 1;
      float o0 = y0m * sWv0[d0] + y1m * sWv1[d0];
      float o1 = y0m * sWv0[d1] + y1m * sWv1[d1];
      float t0 = (o0 - mum) * rsm * sg[d0] + sb[d0];
      float t1 = (o1 - mum) * rsm * sg[d1] + sb[d1];
      t0 = (t0 >= 0.f) ? t0 : alpha * t0;
      t1 = (t1 >= 0.f) ? t1 : alpha * t1;
      v2f a;  a.x  = t0;                   a.y  = t1;
      v2f b0; b0.x = mlp_W[d0 * HCH + n0]; b0.y = mlp_W[d1 * HCH + n0];
      v2f b1; b1.x = mlp_W[d0 * HCH + n1]; b1.y = mlp_W[d1 * HCH + n1];
      c0 = wmma_f32(a, b0, c0);
      c1 = wmma_f32(a, b1, c1);
    }
    const int mbase = mt * 16 + ((lane < 16) ? 0 : 8);
    #pragma unroll
    for (int r = 0; r < 8; r++) {
      int mm = nb + mbase + r;
      h_out[mm * HCH + n0] = c0[r] + bias0;
      h_out[mm * HCH + n1] = c1[r] + bias1;
    }
  }
}

// ---------------------------------------------------------------------------
// Kernel 2: per-node inverse L2 norm of visual rows (wave per node).
// ---------------------------------------------------------------------------
template<int VD>
__global__ __launch_bounds__(256)
void rnorm_kernel(const float* __restrict__ vis, float* __restrict__ rn) {
  int node = (blockIdx.x * blockDim.x + threadIdx.x) >> 5;
  int lane = threadIdx.x & 31;
  if (node >= NTOT) return;
  const float4* row = (const float4*)(vis + (size_t)node * VD);
  float acc = 0.f;
  #pragma unroll 4
  for (int j = lane; j < VD / 4; j += 32) {
    float4 v = row[j];
    acc += v.x * v.x + v.y * v.y + v.z * v.z + v.w * v.w;
  }
  acc = wave_reduce_sum(acc);
  if (lane == 0) rn[node] = rsqrtf(acc + 1e-8f);
}

// ---------------------------------------------------------------------------
// Kernel 3: hlin = h @ W + b for body/face (fp32 WMMA, wave per 16-row tile).
// ---------------------------------------------------------------------------
__global__ __launch_bounds__(256)
void hlin_kernel(const float* __restrict__ h,
                 const float* __restrict__ bW, const float* __restrict__ bB,
                 const float* __restrict__ fW, const float* __restrict__ fB,
                 float* __restrict__ lb, float* __restrict__ lf) {
  int wave = (blockIdx.x * blockDim.x + threadIdx.x) >> 5;
  int lane = threadIdx.x & 31;
  if (wave >= NTOT / 16) return;
  const int ksel = (lane < 16) ? 0 : 2;
  const int m = wave * 16 + (lane & 15);
  const int n0 = lane & 15, n1 = n0 + 16;
  v8f cb0 = {}, cb1 = {}, cf0 = {}, cf1 = {};
  #pragma unroll
  for (int k = 0; k < HCH; k += 4) {
    int d0 = k + ksel, d1 = d0 + 1;
    v2f a;   a.x   = h[m * HCH + d0];     a.y   = h[m * HCH + d1];
    v2f bb0; bb0.x = bW[d0 * HCH + n0];   bb0.y = bW[d1 * HCH + n0];
    v2f bb1; bb1.x = bW[d0 * HCH + n1];   bb1.y = bW[d1 * HCH + n1];
    v2f bf0; bf0.x = fW[d0 * HCH + n0];   bf0.y = fW[d1 * HCH + n0];
    v2f bf1; bf1.x = fW[d0 * HCH + n1];   bf1.y = fW[d1 * HCH + n1];
    cb0 = wmma_f32(a, bb0, cb0);
    cb1 = wmma_f32(a, bb1, cb1);
    cf0 = wmma_f32(a, bf0, cf0);
    cf1 = wmma_f32(a, bf1, cf1);
  }
  const float bb0s = bB[n0], bb1s = bB[n1], fb0s = fB[n0], fb1s = fB[n1];
  const int mbase = wave * 16 + ((lane < 16) ? 0 : 8);
  #pragma unroll
  for (int r = 0; r < 8; r++) {
    int mm = mbase + r;
    lb[mm * HCH + n0] = cb0[r] + bb0s;
    lb[mm * HCH + n1] = cb1[r] + bb1s;
    lf[mm * HCH + n0] = cf0[r] + fb0s;
    lf[mm * HCH + n1] = cf1[r] + fb1s;
  }
}

// ---------------------------------------------------------------------------
// Kernel 4: one wave per edge: cosine weight (dot * rn_src * rn_dst from
// L2-resident visual rows), weighted message scatter via f32 atomics.
// ---------------------------------------------------------------------------
template<int VD>
__global__ __launch_bounds__(256)
void edge_kernel(const int* __restrict__ ei, const float* __restrict__ vis,
                 const float* __restrict__ rn, const float* __restrict__ hlin,
                 float* __restrict__ accum) {
  int e = (blockIdx.x * blockDim.x + threadIdx.x) >> 5;
  int lane = threadIdx.x & 31;
  if (e >= EDGES) return;
  int src = ei[e], dst = ei[EDGES + e];
  const float4* rs = (const float4*)(vis + (size_t)src * VD);
  const float4* rd = (const float4*)(vis + (size_t)dst * VD);
  float acc = 0.f;
  #pragma unroll 4
  for (int j = lane; j < VD / 4; j += 32) {
    float4 a = rs[j], b = rd[j];
    acc += a.x * b.x + a.y * b.y + a.z * b.z + a.w * b.w;
  }
  acc = wave_reduce_sum(acc);
  float w = acc * rn[src] * rn[dst];
  atomicAdd(&accum[(size_t)dst * HCH + lane], w * hlin[(size_t)src * HCH + lane]);
}

// ---------------------------------------------------------------------------
// Kernel 5: scores = h@np_W + hb@pb_W + hf@pf_W + biases (wave per node).
// ---------------------------------------------------------------------------
__global__ __launch_bounds__(256)
void final_kernel(const float* __restrict__ h, const float* __restrict__ hb,
                  const float* __restrict__ hf,
                  const float* __restrict__ npW, const float* __restrict__ npb,
                  const float* __restrict__ pbW, const float* __restrict__ pbb,
                  const float* __restrict__ pfW, const float* __restrict__ pfb,
                  float* __restrict__ out) {
  int i = (blockIdx.x * blockDim.x + threadIdx.x) >> 5;
  int lane = threadIdx.x & 31;
  if (i >= NTOT) return;
  float v = h[i * HCH + lane] * npW[lane]
          + hb[i * HCH + lane] * pbW[lane]
          + hf[i * HCH + lane] * pfW[lane];
  v = wave_reduce_sum(v);
  if (lane == 0) out[i] = v + npb[0] + pbb[0] + pfb[0];
}

extern "C" void kernel_launch(void* const* d_in, const int* in_sizes, int n_in,
                              void* d_out, int out_size, void* d_ws, size_t ws_size,
                              hipStream_t stream) {
  const float* x        = (const float*)d_in[0];
  const float* vbody    = (const float*)d_in[1];
  const float* vface    = (const float*)d_in[2];
  const float* Wq       = (const float*)d_in[3];
  const float* Wk       = (const float*)d_in[4];
  const float* Wv       = (const float*)d_in[5];
  const float* ln_g     = (const float*)d_in[6];
  const float* ln_b     = (const float*)d_in[7];
  const float* prelu_a  = (const float*)d_in[8];
  const float* mlp_W    = (const float*)d_in[9];
  const float* mlp_b    = (const float*)d_in[10];
  const float* np_W     = (const float*)d_in[11];
  const float* np_b     = (const float*)d_in[12];
  const float* body_W   = (const float*)d_in[13];
  const float* body_b   = (const float*)d_in[14];
  const float* face_W   = (const float*)d_in[15];
  const float* face_b   = (const float*)d_in[16];
  const float* pb_W     = (const float*)d_in[17];
  const float* pb_b     = (const float*)d_in[18];
  const float* pf_W     = (const float*)d_in[19];
  const float* pf_b     = (const float*)d_in[20];
  const int*   ei_body  = (const int*)d_in[21];
  const int*   ei_face  = (const int*)d_in[22];
  // d_in[23] = batch_vec (int64) unused: fixed 32 graphs x 256 nodes.

  float* ws = (float*)d_ws;
  float* h  = ws;                             // N*32
  float* lb = h  + (size_t)NTOT * HCH;        // N*32
  float* lf = lb + (size_t)NTOT * HCH;        // N*32
  float* ab = lf + (size_t)NTOT * HCH;        // N*32 (body accum)
  float* af = ab + (size_t)NTOT * HCH;        // N*32 (face accum)
  float* rb = af + (size_t)NTOT * HCH;        // N
  float* rf = rb + NTOT;                      // N

  hipMemsetAsync(ab, 0, 2 * (size_t)NTOT * HCH * sizeof(float), stream);

  attn_mlp_kernel<<<NGRAPH, 256, 0, stream>>>(x, Wq, Wk, Wv, ln_g, ln_b,
                                              prelu_a, mlp_W, mlp_b, h);
  rnorm_kernel<VBODY><<<NTOT / 8, 256, 0, stream>>>(vbody, rb);
  rnorm_kernel<VFACE><<<NTOT / 8, 256, 0, stream>>>(vface, rf);
  hlin_kernel<<<(NTOT / 16) / 8, 256, 0, stream>>>(h, body_W, body_b,
                                                   face_W, face_b, lb, lf);
  edge_kernel<VBODY><<<EDGES / 8, 256, 0, stream>>>(ei_body, vbody, rb, lb, ab);
  edge_kernel<VFACE><<<EDGES / 8, 256, 0, stream>>>(ei_face, vface, rf, lf, af);
  final_kernel<<<NTOT / 8, 256, 0, stream>>>(h, ab, af, np_W, np_b,
                                             pb_W, pb_b, pf_W, pf_b,
                                             (float*)d_out);
}